// SMPLModel_3882650436507
// MI455X (gfx1250) — compile-verified
//
#include <hip/hip_runtime.h>

typedef __attribute__((ext_vector_type(2))) float v2f;
typedef __attribute__((ext_vector_type(8))) float v8f;

constexpr int NB   = 512;          // batch
constexpr int NJ   = 24;           // joints
constexpr int NV   = 6890;         // vertices
constexpr int NC3  = NV * 3;       // 20670 (vertex*xyz flat)
constexpr int KDIM = 207;          // pose-blend inner dim
constexpr int KP   = 208;          // padded K (zero tail)
constexpr int RES_SIZE = NB * NC3; // result elements before joints

__constant__ int c_parents[NJ] = {0,0,0,0,1,2,3,4,5,6,7,8,9,9,9,12,13,14,16,17,18,19,20,21};

__device__ __forceinline__ int imin(int a, int b) { return a < b ? a : b; }

// ---------------------------------------------------------------------------
// 1) v_shaped[n, i] = v_template[i] + sum_k betas[n,k] * shapedirs[i,k]
// ---------------------------------------------------------------------------
__global__ void k_shape_blend(const float* __restrict__ betas,
                              const float* __restrict__ v_template,
                              const float* __restrict__ shapedirs,
                              float* __restrict__ vsh) {
    int idx = blockIdx.x * blockDim.x + threadIdx.x;
    if (idx >= NB * NC3) return;
    int n = idx / NC3;
    int i = idx - n * NC3;
    const float* b = betas + n * 10;
    const float* s = shapedirs + i * 10;
    float acc = v_template[i];
#pragma unroll
    for (int k = 0; k < 10; ++k) acc += b[k] * s[k];
    vsh[idx] = acc;
}

// ---------------------------------------------------------------------------
// 2) J[n,j,c] = sum_v J_regressor[j,v] * v_shaped[n,v,c]   (block per (j,n))
// ---------------------------------------------------------------------------
__global__ void k_joint_regress(const float* __restrict__ Jreg,   // [24,6890]
                                const float* __restrict__ vsh,    // [512,20670]
                                float* __restrict__ Jbuf) {       // [512,24,3]
    int j = blockIdx.x, n = blockIdx.y, t = threadIdx.x;
    __shared__ float red[256 * 3];
    float s0 = 0.f, s1 = 0.f, s2 = 0.f;
    const float* w  = Jreg + j * NV;
    const float* vb = vsh + n * NC3;
    for (int v = t; v < NV; v += 256) {
        float ww = w[v];
        s0 += ww * vb[v * 3 + 0];
        s1 += ww * vb[v * 3 + 1];
        s2 += ww * vb[v * 3 + 2];
    }
    red[t] = s0; red[256 + t] = s1; red[512 + t] = s2;
    __syncthreads();
    for (int off = 128; off > 0; off >>= 1) {
        if (t < off) {
            red[t]       += red[t + off];
            red[256 + t] += red[256 + t + off];
            red[512 + t] += red[512 + t + off];
        }
        __syncthreads();
    }
    if (t == 0) {
        float* o = Jbuf + (n * NJ + j) * 3;
        o[0] = red[0]; o[1] = red[256]; o[2] = red[512];
    }
}

// ---------------------------------------------------------------------------
// 3) Per-body: Rodrigues, lrotmin, FK chain, rest-pose removal.
//    One thread per body (512 total) — negligible work, keep it simple.
// ---------------------------------------------------------------------------
__global__ void k_fk(const float* __restrict__ thetas, // [512,72]
                     const float* __restrict__ Jbuf,   // [512,24,3]
                     float* __restrict__ Gbuf,         // [512,24,12] row-major 3x4
                     float* __restrict__ lrot) {       // [512,208] zero-padded
    int n = blockIdx.x * blockDim.x + threadIdx.x;
    if (n >= NB) return;

    float R[NJ][9];
    for (int j = 0; j < NJ; ++j) {
        float rx = thetas[n * 72 + j * 3 + 0];
        float ry = thetas[n * 72 + j * 3 + 1];
        float rz = thetas[n * 72 + j * 3 + 2];
        float th = sqrtf(rx * rx + ry * ry + rz * rz) + 1e-8f;
        float inv = 1.0f / th;
        float ux = rx * inv, uy = ry * inv, uz = rz * inv;
        float c = cosf(th), s = sinf(th), mc = 1.0f - c;
        R[j][0] = c + mc * ux * ux;     R[j][1] = mc * ux * uy - s * uz; R[j][2] = mc * ux * uz + s * uy;
        R[j][3] = mc * uy * ux + s * uz; R[j][4] = c + mc * uy * uy;     R[j][5] = mc * uy * uz - s * ux;
        R[j][6] = mc * uz * ux - s * uy; R[j][7] = mc * uz * uy + s * ux; R[j][8] = c + mc * uz * uz;
    }

    // lrotmin = (R[1:] - I).flatten(), padded to 208 with one zero
    float* lr = lrot + n * KP;
    for (int j = 1; j < NJ; ++j)
        for (int e = 0; e < 9; ++e)
            lr[(j - 1) * 9 + e] = R[j][e] - ((e == 0 || e == 4 || e == 8) ? 1.0f : 0.0f);
    lr[207] = 0.0f;

    // forward kinematics (3x4 affine composition, implicit [0 0 0 1] row)
    float GR[NJ][9], GT[NJ][3];
    const float* J = Jbuf + n * NJ * 3;
    for (int e = 0; e < 9; ++e) GR[0][e] = R[0][e];
    GT[0][0] = J[0]; GT[0][1] = J[1]; GT[0][2] = J[2];
    for (int i = 1; i < NJ; ++i) {
        int p = c_parents[i];
        float tx = J[i * 3 + 0] - J[p * 3 + 0];
        float ty = J[i * 3 + 1] - J[p * 3 + 1];
        float tz = J[i * 3 + 2] - J[p * 3 + 2];
        for (int r = 0; r < 3; ++r) {
            for (int cc = 0; cc < 3; ++cc)
                GR[i][r * 3 + cc] = GR[p][r * 3 + 0] * R[i][0 + cc]
                                  + GR[p][r * 3 + 1] * R[i][3 + cc]
                                  + GR[p][r * 3 + 2] * R[i][6 + cc];
            GT[i][r] = GR[p][r * 3 + 0] * tx + GR[p][r * 3 + 1] * ty
                     + GR[p][r * 3 + 2] * tz + GT[p][r];
        }
    }

    // remove rest-pose joint locations: t' = t - R*J_i ; store [R | t'] (3x4)
    for (int i = 0; i < NJ; ++i) {
        float jx = J[i * 3 + 0], jy = J[i * 3 + 1], jz = J[i * 3 + 2];
        float* g = Gbuf + (n * NJ + i) * 12;
        for (int r = 0; r < 3; ++r) {
            g[r * 4 + 0] = GR[i][r * 3 + 0];
            g[r * 4 + 1] = GR[i][r * 3 + 1];
            g[r * 4 + 2] = GR[i][r * 3 + 2];
            g[r * 4 + 3] = GT[i][r] - (GR[i][r * 3 + 0] * jx + GR[i][r * 3 + 1] * jy + GR[i][r * 3 + 2] * jz);
        }
    }
}

// ---------------------------------------------------------------------------
// 4) Pose blend GEMM via V_WMMA_F32_16X16X4_F32:
//    vsh[n, i] += sum_p lrot[n, p] * posedirs[i, p]
//    M=512 (n), N=20672 (i, padded), K=207 (p, padded to 208 on the A side).
//    Wave computes one 16x64 tile: A fragment reused by 4 WMMAs per k-step.
// ---------------------------------------------------------------------------
__global__ void k_pose_blend_wmma(const float* __restrict__ posedirs, // [20670,207]
                                  const float* __restrict__ lrot,     // [512,208]
                                  float* __restrict__ vsh) {          // [512,20670] in/out
    int wave = blockIdx.x * (blockDim.x >> 5) + (threadIdx.x >> 5);
    int lane = threadIdx.x & 31;
    int mt   = wave & 31;      // 32 M tiles of 16
    int nc   = wave >> 5;      // 323 N chunks of 64
    int m0   = mt << 4;
    int n0   = nc << 6;
    int half = lane >> 4;      // 0: K pair {0,1}, 1: K pair {2,3}
    int l16  = lane & 15;
    int half2 = half << 1;

    // A: lane l holds lrot row (m0 + l%16), K slots half2, half2+1 (8B aligned)
    const float* arow = lrot + (m0 + l16) * KP + half2;

    // B: lane l holds posedirs row for column n-sub + l%16 (clamped for N pad)
    const float* pr[4];
#pragma unroll
    for (int nt = 0; nt < 4; ++nt) {
        int col = imin(n0 + (nt << 4) + l16, NC3 - 1);
        pr[nt] = posedirs + col * KDIM;
    }

    v8f acc[4] = {};
#pragma unroll 2
    for (int p = 0; p < KDIM; p += 4) {   // 52 k-steps
        v2f a = *(const v2f*)(arow + p);  // lrot zero-padded at index 207
        int ka = p + half2;               // <= 206
        int kb = imin(ka + 1, KDIM - 1);  // clamp 207 -> 206 (A pad is 0 there)
#pragma unroll
        for (int nt = 0; nt < 4; ++nt) {
            v2f b;
            b.x = pr[nt][ka];
            b.y = pr[nt][kb];
            acc[nt] = __builtin_amdgcn_wmma_f32_16x16x4_f32(
                false, a, false, b, (short)0, acc[nt], false, false);
        }
    }

    // D layout: VGPR r, lane l -> (M = r + (l>=16)*8, N = l%16)
    int mbase = m0 + (half << 3);
#pragma unroll
    for (int nt = 0; nt < 4; ++nt) {
        int col = n0 + (nt << 4) + l16;
        if (col < NC3) {
#pragma unroll
            for (int r = 0; r < 8; ++r)
                vsh[(mbase + r) * NC3 + col] += acc[nt][r];
        }
    }
}

// ---------------------------------------------------------------------------
// 5) Linear blend skinning. Block handles 256 vertices of one body; the
//    body's 24 3x4 transforms are staged in LDS.
// ---------------------------------------------------------------------------
__global__ void k_skin(const float* __restrict__ weights, // [6890,24]
                       const float* __restrict__ Gbuf,    // [512,24,12]
                       const float* __restrict__ vsh,     // v_posed [512,20670]
                       const float* __restrict__ trans,   // [512,3]
                       float* __restrict__ out) {         // [512,6890,3]
    int n = blockIdx.y;
    int t = threadIdx.x;
    __shared__ float sG[NJ * 12];
    __shared__ float sT[3];
    for (int e = t; e < NJ * 12; e += 256) sG[e] = Gbuf[n * NJ * 12 + e];
    if (t < 3) sT[t] = trans[n * 3 + t];
    __syncthreads();

    int v = blockIdx.x * blockDim.x + t;
    if (v >= NV) return;

    float T[12];
#pragma unroll
    for (int e = 0; e < 12; ++e) T[e] = 0.f;
    const float* w = weights + v * NJ;
    for (int j = 0; j < NJ; ++j) {
        float wj = w[j];
#pragma unroll
        for (int e = 0; e < 12; ++e) T[e] += wj * sG[j * 12 + e];
    }
    const float* p = vsh + n * NC3 + v * 3;
    float px = p[0], py = p[1], pz = p[2];
    float* o = out + n * NC3 + v * 3;
    o[0] = T[0] * px + T[1] * py + T[2]  * pz + T[3]  + sT[0];
    o[1] = T[4] * px + T[5] * py + T[6]  * pz + T[7]  + sT[1];
    o[2] = T[8] * px + T[9] * py + T[10] * pz + T[11] + sT[2];
}

// ---------------------------------------------------------------------------
// 6) joints[n,j,c] = sum_v result[n,v,c] * joint_regressor[v,j]
// ---------------------------------------------------------------------------
__global__ void k_joints(const float* __restrict__ jreg, // [6890,24]
                         const float* __restrict__ res,  // [512,6890,3]
                         float* __restrict__ jout) {     // [512,24,3]
    int j = blockIdx.x, n = blockIdx.y, t = threadIdx.x;
    __shared__ float red[256 * 3];
    float s0 = 0.f, s1 = 0.f, s2 = 0.f;
    const float* r = res + n * NC3;
    for (int v = t; v < NV; v += 256) {
        float w = jreg[v * NJ + j];
        s0 += w * r[v * 3 + 0];
        s1 += w * r[v * 3 + 1];
        s2 += w * r[v * 3 + 2];
    }
    red[t] = s0; red[256 + t] = s1; red[512 + t] = s2;
    __syncthreads();
    for (int off = 128; off > 0; off >>= 1) {
        if (t < off) {
            red[t]       += red[t + off];
            red[256 + t] += red[256 + t + off];
            red[512 + t] += red[512 + t + off];
        }
        __syncthreads();
    }
    if (t == 0) {
        float* o = jout + (n * NJ + j) * 3;
        o[0] = red[0]; o[1] = red[256]; o[2] = red[512];
    }
}

// ---------------------------------------------------------------------------
extern "C" void kernel_launch(void* const* d_in, const int* in_sizes, int n_in,
                              void* d_out, int out_size, void* d_ws, size_t ws_size,
                              hipStream_t stream) {
    const float* betas      = (const float*)d_in[0];
    const float* thetas     = (const float*)d_in[1];
    const float* trans      = (const float*)d_in[2];
    const float* v_template = (const float*)d_in[3];
    const float* shapedirs  = (const float*)d_in[4];
    const float* posedirs   = (const float*)d_in[5];
    const float* Jreg       = (const float*)d_in[6];
    const float* jointreg   = (const float*)d_in[7];
    const float* weights    = (const float*)d_in[8];
    float* out = (float*)d_out;

    float* ws   = (float*)d_ws;
    float* vsh  = ws;                       // 512*20670 floats (v_shaped -> v_posed)
    float* Jbuf = vsh + NB * NC3;           // 512*72
    float* Gbuf = Jbuf + NB * NJ * 3;       // 512*288
    float* lrot = Gbuf + NB * NJ * 12;      // 512*208

    k_shape_blend<<<(NB * NC3 + 255) / 256, 256, 0, stream>>>(betas, v_template, shapedirs, vsh);
    k_joint_regress<<<dim3(NJ, NB), 256, 0, stream>>>(Jreg, vsh, Jbuf);
    k_fk<<<(NB + 63) / 64, 64, 0, stream>>>(thetas, Jbuf, Gbuf, lrot);
    // 32 M-tiles * 323 N-chunks = 10336 wave jobs; 8 waves/block -> 1292 blocks
    k_pose_blend_wmma<<<1292, 256, 0, stream>>>(posedirs, lrot, vsh);
    k_skin<<<dim3((NV + 255) / 256, NB), 256, 0, stream>>>(weights, Gbuf, vsh, trans, out);
    k_joints<<<dim3(NJ, NB), 256, 0, stream>>>(jointreg, out, out + RES_SIZE);
}